// OptimizedResidualVectorQuantizer_76544907149322
// MI455X (gfx1250) — compile-verified
//
#include <hip/hip_runtime.h>
#include <hip/hip_bf16.h>

typedef float v2f __attribute__((ext_vector_type(2)));
typedef float v4f __attribute__((ext_vector_type(4)));
typedef float v8f __attribute__((ext_vector_type(8)));

#define N_ROWS       16384   // B*S
#define DIM          512
#define QSTAGES      8
#define KCODES       8192
#define ROWS_PER_BLK 64      // 4 WMMA M-tiles per block
#define NJ           (DIM / 4)   // 128 K-steps of 4
#define LSTR         268     // dwords per (mt,lane) strip: 16B aligned, 2-cycle b128

// ---------------------------------------------------------------------------
// Kernel 1: squared norms of every codebook row (one wave32 per row)
// ---------------------------------------------------------------------------
__global__ void rvq_cnorm_kernel(const float* __restrict__ cb,
                                 float* __restrict__ cnorm) {
    int wid  = (blockIdx.x * blockDim.x + threadIdx.x) >> 5;
    int lane = threadIdx.x & 31;
    if (wid >= QSTAGES * KCODES) return;
    const float* row = cb + (size_t)wid * DIM;
    float s = 0.f;
    #pragma unroll
    for (int c = lane * 4; c < DIM; c += 32 * 4) {
        v4f t = *(const v4f*)(row + c);
        s += t.x * t.x + t.y * t.y + t.z * t.z + t.w * t.w;
    }
    #pragma unroll
    for (int off = 16; off > 0; off >>= 1) s += __shfl_xor(s, off, 32);
    if (lane == 0) cnorm[wid] = s;
}

// ---------------------------------------------------------------------------
// Kernel 2: ALL 8 RVQ stages fused.  Block = 256 threads (8 waves) owns 64
// rows; the residual tile lives in LDS (WMMA-fragment-swizzled) for the whole
// algorithm and is updated in place per stage.  Only x, the codebooks, the
// quantized output and the indices touch global memory.
//
// LDS A-tile swizzle:
//   sA[(mt*32 + lane)*LSTR + 2*j + sub]  =  residual[mt*16 + (lane&15)]
//                                                  [4*j + 2*(lane>>4) + sub]
// so one ds_load_b128 per (mt, 2 K-steps).
// ---------------------------------------------------------------------------
__global__ __launch_bounds__(256)
void rvq_fused_kernel(const float* __restrict__ x,      // [N,D] input
                      const float* __restrict__ cb,     // [Q,K,D] codebooks
                      const float* __restrict__ cnorm,  // [Q,K] ||e||^2
                      float*       __restrict__ quant,  // [N,D] output
                      float*       __restrict__ idxOut, // [N,Q] indices (as f32)
                      float*       __restrict__ lossOut) {
    extern __shared__ float smem[];
    float* sA   = smem;                                  // 4*32*LSTR floats
    float* redV = smem + 4 * 32 * LSTR;                  // 8*64
    int*   redI = (int*)(redV + 8 * ROWS_PER_BLK);       // 8*64
    int*   sIdx = redI + 8 * ROWS_PER_BLK;               // 64

    const int tid     = threadIdx.x;
    const int wave    = tid >> 5;
    const int lane    = tid & 31;
    const int rowBase = blockIdx.x * ROWS_PER_BLK;

    // ---- stage x tile into LDS, swizzled to fragment order ----
    for (int i = tid; i < ROWS_PER_BLK * (DIM / 4); i += 256) {
        int r = i >> 7;                 // row 0..63
        int c = (i & 127) * 4;          // col 0..508 step 4
        v4f t = *(const v4f*)(x + (size_t)(rowBase + r) * DIM + c);
        int mt = r >> 4, nn = r & 15, j2 = (c >> 2) * 2;  // slot = 2*(c/4)
        v2f lo; lo.x = t.x; lo.y = t.y;                   // koff 0 half (lane n)
        v2f hi; hi.x = t.z; hi.y = t.w;                   // koff 2 half (lane n+16)
        *(v2f*)(sA + (mt * 32 + nn)      * LSTR + j2) = lo;
        *(v2f*)(sA + (mt * 32 + nn + 16) * LSTR + j2) = hi;
    }
    __syncthreads();

    // WMMA fp32 16x16x4 fragment coordinates for this lane
    const int n    = lane & 15;          // column / N index
    const int koff = (lane >> 4) * 2;    // K sub-offset (0 or 2)

    int aBase[4];
    #pragma unroll
    for (int mt = 0; mt < 4; ++mt) aBase[mt] = (mt * 32 + lane) * LSTR;

    // ================= stage loop: 8 residual quantizer stages =============
    for (int q = 0; q < QSTAGES; ++q) {
        const float* cbQ = cb    + (size_t)q * KCODES * DIM;
        const float* cnQ = cnorm + (size_t)q * KCODES;

        float bestV[4][8];
        int   bestI[4][8];
        #pragma unroll
        for (int mt = 0; mt < 4; ++mt)
            #pragma unroll
            for (int e = 0; e < 8; ++e) { bestV[mt][e] = 3.4e38f; bestI[mt][e] = 0; }

        // ---- sweep this wave's 1/8 slice of the 8192 codes, 16 at a time ----
        for (int t = 0; t < KCODES / (16 * 8); ++t) {
            const int codeBase = (t * 8 + wave) * 16;
            const float* bPtr = cbQ + (size_t)(codeBase + n) * DIM + koff;
            if (t + 1 < KCODES / (16 * 8)) {
                // pull next tile's codebook rows toward L0/L2 (global_prefetch_b8)
                __builtin_prefetch(cbQ + (size_t)((t + 1) * 8 + wave) * 16 * DIM
                                       + (size_t)n * DIM, 0, 1);
            }

            v8f acc[4] = {v8f{0}, v8f{0}, v8f{0}, v8f{0}};

            #pragma unroll 2
            for (int j = 0; j < NJ; j += 2) {       // two K-steps per iteration
                v2f b0 = *(const v2f*)(bPtr + 4 * j);
                v2f b1 = *(const v2f*)(bPtr + 4 * j + 4);
                v4f ap[4];
                #pragma unroll
                for (int mt = 0; mt < 4; ++mt)
                    ap[mt] = *(const v4f*)(sA + aBase[mt] + 2 * j);  // 16B aligned
                #pragma unroll
                for (int mt = 0; mt < 4; ++mt) {
                    v2f a; a.x = ap[mt].x; a.y = ap[mt].y;
                    acc[mt] = __builtin_amdgcn_wmma_f32_16x16x4_f32(
                        false, a, false, b0, (short)0, acc[mt], false, false);
                }
                #pragma unroll
                for (int mt = 0; mt < 4; ++mt) {
                    v2f a; a.x = ap[mt].z; a.y = ap[mt].w;
                    acc[mt] = __builtin_amdgcn_wmma_f32_16x16x4_f32(
                        false, a, false, b1, (short)0, acc[mt], false, false);
                }
            }

            const float cn = cnQ[codeBase + n];   // same for all 8 acc slots
            #pragma unroll
            for (int mt = 0; mt < 4; ++mt)
                #pragma unroll
                for (int e = 0; e < 8; ++e) {
                    float d2 = cn - 2.0f * acc[mt][e];   // ||e||^2 - 2 r.e
                    if (d2 < bestV[mt][e]) { bestV[mt][e] = d2; bestI[mt][e] = codeBase + n; }
                }
        }

        // ---- butterfly argmin across the 16 N-lanes of each half-wave ----
        #pragma unroll
        for (int mt = 0; mt < 4; ++mt)
            #pragma unroll
            for (int e = 0; e < 8; ++e) {
                float bv = bestV[mt][e];
                int   bi = bestI[mt][e];
                #pragma unroll
                for (int off = 1; off < 16; off <<= 1) {
                    float ov = __shfl_xor(bv, off, 32);
                    int   oi = __shfl_xor(bi, off, 32);
                    if (ov < bv || (ov == bv && oi < bi)) { bv = ov; bi = oi; }
                }
                if (n == 0) {
                    int row = mt * 16 + e + 8 * (lane >> 4);  // C-layout row here
                    redV[wave * ROWS_PER_BLK + row] = bv;
                    redI[wave * ROWS_PER_BLK + row] = bi;
                }
            }
        __syncthreads();

        // ---- cross-wave argmin; emit index ----
        if (tid < ROWS_PER_BLK) {
            float bv = redV[tid];
            int   bi = redI[tid];
            #pragma unroll
            for (int w = 1; w < 8; ++w) {
                float ov = redV[w * ROWS_PER_BLK + tid];
                int   oi = redI[w * ROWS_PER_BLK + tid];
                if (ov < bv || (ov == bv && oi < bi)) { bv = ov; bi = oi; }
            }
            sIdx[tid] = bi;
            idxOut[(size_t)(rowBase + tid) * QSTAGES + q] = (float)bi;
        }
        __syncthreads();

        // ---- residual -= cb[idx], in place in the swizzled LDS tile ----
        for (int i = tid; i < ROWS_PER_BLK * (DIM / 4); i += 256) {
            int r = i >> 7;
            int c = (i & 127) * 4;
            int mt = r >> 4, nn = r & 15, j2 = (c >> 2) * 2;
            const float* crow = cbQ + (size_t)sIdx[r] * DIM + c;
            float* pLo = sA + (mt * 32 + nn)      * LSTR + j2;
            float* pHi = sA + (mt * 32 + nn + 16) * LSTR + j2;
            v2f lo = *(v2f*)pLo, hi = *(v2f*)pHi;
            lo.x -= crow[0]; lo.y -= crow[1];
            hi.x -= crow[2]; hi.y -= crow[3];
            *(v2f*)pLo = lo;
            *(v2f*)pHi = hi;
        }
        __syncthreads();
    }

    // ---- epilogue: quantized = x - residual_final (x re-read, L2-warm) ----
    for (int i = tid; i < ROWS_PER_BLK * (DIM / 4); i += 256) {
        int r = i >> 7;
        int c = (i & 127) * 4;
        int mt = r >> 4, nn = r & 15, j2 = (c >> 2) * 2;
        v2f lo = *(const v2f*)(sA + (mt * 32 + nn)      * LSTR + j2);
        v2f hi = *(const v2f*)(sA + (mt * 32 + nn + 16) * LSTR + j2);
        v4f xv = *(const v4f*)(x + (size_t)(rowBase + r) * DIM + c);
        v4f o;
        o.x = xv.x - lo.x;
        o.y = xv.y - lo.y;
        o.z = xv.z - hi.x;
        o.w = xv.w - hi.y;
        *(v4f*)(quant + (size_t)(rowBase + r) * DIM + c) = o;
    }
    if (blockIdx.x == 0 && tid == 0) *lossOut = 0.0f;
}

// ---------------------------------------------------------------------------
extern "C" void kernel_launch(void* const* d_in, const int* in_sizes, int n_in,
                              void* d_out, int out_size, void* d_ws, size_t ws_size,
                              hipStream_t stream) {
    const float* x  = (const float*)d_in[0];   // [B,S,D] f32
    const float* cb = (const float*)d_in[1];   // [Q,K,D] f32

    float* quant   = (float*)d_out;                          // N*D
    float* idxOut  = quant + (size_t)N_ROWS * DIM;           // N*Q (indices as f32)
    float* lossOut = idxOut + (size_t)N_ROWS * QSTAGES;      // 1

    float* cnorm = (float*)d_ws;                             // Q*K norms

    // 1) codebook squared norms (one wave per row, 8 waves per block)
    rvq_cnorm_kernel<<<(QSTAGES * KCODES) / 8, 256, 0, stream>>>(cb, cnorm);

    // 2) all 8 stages fused; residual tile resident in LDS
    const size_t smemBytes =
        (4 * 32 * LSTR + 8 * ROWS_PER_BLK) * sizeof(float) +
        8 * ROWS_PER_BLK * sizeof(int) + ROWS_PER_BLK * sizeof(int);
    rvq_fused_kernel<<<N_ROWS / ROWS_PER_BLK, 256, smemBytes, stream>>>(
        x, cb, cnorm, quant, idxOut, lossOut);
}